// NMMR_Q_Loss_3891240370302
// MI455X (gfx1250) — compile-verified
//
#include <hip/hip_runtime.h>

// ---------------------------------------------------------------------------
// NMMR Q-loss on MI455X (gfx1250, wave32).
//   wx = [w|x] : 12288 x 16 fp32 (L2-resident, ~884KB total inputs)
//   loss = sum_g [ v_gT exp(-(d2)) v_g / Na_g^2 ],  d2 = sq_i+sq_j-2 wx_i.wx_j
// Gram tiles via V_WMMA_F32_16X16X4_F32 (4 chained, K=16); epilogue is
// v_exp_f32-bound. Deterministic: no float atomics, fixed reduction order.
// ---------------------------------------------------------------------------

#define N_ROWS 12288
#define TILE 16
#define WAVES_PER_BLOCK 8
#define GRID_X 96             // 96 * 8 waves * 16 rows = 12288
#define GRID_Y 24
#define JTILES_PER_CHUNK 32   // 24 * 32 * 16 cols = 12288
#define NBLOCKS (GRID_X * GRID_Y)
#define GAMMA_LOG2E 1.4426950408889634f   // GAMMA * log2(e), GAMMA = 1.0

typedef float v2f __attribute__((ext_vector_type(2)));
typedef float v8f __attribute__((ext_vector_type(8)));

// One aligned float2 from the virtual concatenated row [w|x]; K-pairs start at
// even k so they never straddle the w/x boundary.
__device__ __forceinline__ v2f load_pair(const float* __restrict__ w,
                                         const float* __restrict__ x,
                                         int row, int k) {
  const float* p = (k < 8) ? (w + row * 8 + k) : (x + row * 8 + (k - 8));
  return *(const v2f*)p;
}

// ---- pass 1: per-row sq, group-masked (q-1) vectors --------------------------
__global__ void precompute_kernel(const float* __restrict__ q,
                                  const float* __restrict__ a,
                                  const float* __restrict__ w,
                                  const float* __restrict__ x,
                                  float* __restrict__ sq,
                                  float* __restrict__ g0,
                                  float* __restrict__ g1) {
  const int i = blockIdx.x * blockDim.x + threadIdx.x;
  if (i >= N_ROWS) return;
  float s = 0.0f;
#pragma unroll
  for (int k = 0; k < 8; ++k) { const float v = w[i * 8 + k]; s += v * v; }
#pragma unroll
  for (int k = 0; k < 8; ++k) { const float v = x[i * 8 + k]; s += v * v; }
  sq[i] = s;
  const float qm1 = q[i] - 1.0f;
  const float av  = a[i];
  g0[i] = (av == 0.0f) ? qm1 : 0.0f;
  g1[i] = (av == 1.0f) ? qm1 : 0.0f;
}

// ---- pass 2: tiled pairwise kernel with WMMA Gram tiles ----------------------
__global__ void __launch_bounds__(256) pair_kernel(
    const float* __restrict__ w,  const float* __restrict__ x,
    const float* __restrict__ sq, const float* __restrict__ g0,
    const float* __restrict__ g1, float* __restrict__ partials) {
  const int lane = threadIdx.x & 31;
  const int wave = threadIdx.x >> 5;
  const int hi   = lane >> 4;     // lane half: selects K sub-pair / M half
  const int lo   = lane & 15;

  const int i0 = (blockIdx.x * WAVES_PER_BLOCK + wave) * TILE;

  // A operand for this wave's 16-row strip (reused across all 32 j-tiles).
  // Layout (16x16x4 f32): lane<16 holds row=lane, K={k0,k0+1};
  //                       lane>=16 holds row=lane-16, K={k0+2,k0+3}.
  v2f aop[4];
#pragma unroll
  for (int s = 0; s < 4; ++s)
    aop[s] = load_pair(w, x, i0 + lo, 4 * s + 2 * hi);

  // Row-side scalars for the C/D elements this lane owns: M = r + 8*hi.
  float sqi[8], gi0[8], gi1[8];
  const int ibase = i0 + 8 * hi;
#pragma unroll
  for (int r = 0; r < 8; ++r) {
    sqi[r] = sq[ibase + r];
    gi0[r] = g0[ibase + r];
    gi1[r] = g1[ibase + r];
  }

  float acc0 = 0.0f, acc1 = 0.0f;

  const int jt0 = blockIdx.y * JTILES_PER_CHUNK;
  for (int t = 0; t < JTILES_PER_CHUNK; ++t) {
    const int j0   = (jt0 + t) * TILE;
    const int jrow = j0 + lo;

    // B operand (4x16 per step): same per-lane pattern as A, column = jrow.
    v2f bop[4];
#pragma unroll
    for (int s = 0; s < 4; ++s)
      bop[s] = load_pair(w, x, jrow, 4 * s + 2 * hi);

    const float sqj = sq[jrow];
    const float gj0 = g0[jrow];
    const float gj1 = g1[jrow];

    // 16x16 Gram tile: 4 chained f32 WMMAs cover K=16.
    v8f c = {0.f, 0.f, 0.f, 0.f, 0.f, 0.f, 0.f, 0.f};
#pragma unroll
    for (int s = 0; s < 4; ++s)
      c = __builtin_amdgcn_wmma_f32_16x16x4_f32(
          /*neg_a=*/false, aop[s], /*neg_b=*/false, bop[s],
          /*c_mod=*/(short)0, c, /*reuse_a=*/false, /*reuse_b=*/false);

    // Epilogue: d2 -> exp -> two masked accumulations. v_exp_f32 bound.
#pragma unroll
    for (int r = 0; r < 8; ++r) {
      float d2 = sqi[r] + sqj - 2.0f * c[r];
      d2 = fmaxf(d2, 0.0f);
      const float e = __builtin_amdgcn_exp2f(-GAMMA_LOG2E * d2);
      acc0 = fmaf(e, gi0[r] * gj0, acc0);
      acc1 = fmaf(e, gi1[r] * gj1, acc1);
    }
  }

  // wave32 butterfly reduction
#pragma unroll
  for (int m = 16; m >= 1; m >>= 1) {
    acc0 += __shfl_xor(acc0, m, 32);
    acc1 += __shfl_xor(acc1, m, 32);
  }

  __shared__ float red[WAVES_PER_BLOCK][2];
  if (lane == 0) { red[wave][0] = acc0; red[wave][1] = acc1; }
  __syncthreads();
  if (threadIdx.x == 0) {
    float s0 = 0.0f, s1 = 0.0f;
#pragma unroll
    for (int v = 0; v < WAVES_PER_BLOCK; ++v) { s0 += red[v][0]; s1 += red[v][1]; }
    const int blk = blockIdx.y * GRID_X + blockIdx.x;
    partials[2 * blk + 0] = s0;
    partials[2 * blk + 1] = s1;
  }
}

// ---- pass 3: deterministic final reduction ----------------------------------
__global__ void finalize_kernel(const float* __restrict__ a,
                                const float* __restrict__ partials,
                                float* __restrict__ out) {
  __shared__ float s0s[256], s1s[256], c0s[256], c1s[256];
  const int t = threadIdx.x;
  float s0 = 0.f, s1 = 0.f, c0 = 0.f, c1 = 0.f;
  for (int p = t; p < NBLOCKS; p += 256) {
    s0 += partials[2 * p + 0];
    s1 += partials[2 * p + 1];
  }
  for (int i = t; i < N_ROWS; i += 256) {
    const float av = a[i];
    c0 += (av == 0.0f) ? 1.0f : 0.0f;
    c1 += (av == 1.0f) ? 1.0f : 0.0f;
  }
  s0s[t] = s0; s1s[t] = s1; c0s[t] = c0; c1s[t] = c1;
  __syncthreads();
  for (int step = 128; step >= 1; step >>= 1) {
    if (t < step) {
      s0s[t] += s0s[t + step];
      s1s[t] += s1s[t + step];
      c0s[t] += c0s[t + step];
      c1s[t] += c1s[t + step];
    }
    __syncthreads();
  }
  if (t == 0) {
    const float Na0 = c0s[0], Na1 = c1s[0];
    float loss = 0.0f;
    if (Na0 >= 2.0f) loss += s0s[0] / (Na0 * Na0);
    if (Na1 >= 2.0f) loss += s1s[0] / (Na1 * Na1);
    out[0] = loss;
  }
}

extern "C" void kernel_launch(void* const* d_in, const int* in_sizes, int n_in,
                              void* d_out, int out_size, void* d_ws, size_t ws_size,
                              hipStream_t stream) {
  const float* q = (const float*)d_in[0];   // q_a_hat  (N)
  const float* a = (const float*)d_in[1];   // a        (N)
  const float* w = (const float*)d_in[2];   // w        (N*8)
  const float* x = (const float*)d_in[3];   // x        (N*8)
  float* out = (float*)d_out;

  float* sq       = (float*)d_ws;           // N floats
  float* g0       = sq + N_ROWS;            // N floats
  float* g1       = g0 + N_ROWS;            // N floats
  float* partials = g1 + N_ROWS;            // 2*NBLOCKS floats

  precompute_kernel<<<(N_ROWS + 255) / 256, 256, 0, stream>>>(q, a, w, x, sq, g0, g1);
  pair_kernel<<<dim3(GRID_X, GRID_Y), 256, 0, stream>>>(w, x, sq, g0, g1, partials);
  finalize_kernel<<<1, 256, 0, stream>>>(a, partials, out);
}